// GCN_base_39668317946065
// MI455X (gfx1250) — compile-verified
//
#include <hip/hip_runtime.h>
#include <hip/hip_bf16.h>
#include <math.h>

typedef __attribute__((ext_vector_type(2))) float v2f;
typedef __attribute__((ext_vector_type(8))) float v8f;

#define N_NODES   100000
#define N_EDGES   1600000
#define F_IN      64
#define HIDDEN    64
#define N_CLASSES 2

// Non-returning relaxed agent-scope float atomic -> global_atomic_add_f32 (no CAS loop)
static __device__ __forceinline__ void atomic_add_f32(float* p, float v) {
  __hip_atomic_fetch_add(p, v, __ATOMIC_RELAXED, __HIP_MEMORY_SCOPE_AGENT);
}

// Full-precision f32 WMMA: D = A(16x4) * B(4x16) + C, chained over K.
// Builtin probe must run only in the DEVICE pass; the host pass just needs
// this to parse (it is never executed on the host).
static __device__ __forceinline__ v8f wmma_f32x4(v2f a, v2f b, v8f c) {
#if defined(__HIP_DEVICE_COMPILE__)
#  if __has_builtin(__builtin_amdgcn_wmma_f32_16x16x4_f32)
  return __builtin_amdgcn_wmma_f32_16x16x4_f32(false, a, false, b, (short)0, c,
                                               false, false);
#  else
#    error "DEVICE pass: __builtin_amdgcn_wmma_f32_16x16x4_f32 not declared for gfx1250"
#  endif
#else
  (void)a; (void)b;
  return c;  // host stub, never executed
#endif
}

// ---------------------------------------------------------------------------
// 0) init: deg = 1.0 (self-loop), zero both aggregation buffers (graph-replay safe)
__global__ void init_ws(float* __restrict__ deg, float* __restrict__ agg1,
                        float* __restrict__ agg2) {
  int i = blockIdx.x * blockDim.x + threadIdx.x;
  if (i < N_NODES * HIDDEN)    agg1[i] = 0.0f;
  if (i < N_NODES * N_CLASSES) agg2[i] = 0.0f;
  if (i < N_NODES)             deg[i]  = 1.0f;
}

// 1) degree accumulation over edge targets (row = edge_index[0])
__global__ void degree_kernel(const long long* __restrict__ rows,
                              float* __restrict__ deg) {
  int e = blockIdx.x * blockDim.x + threadIdx.x;
  if (e < N_EDGES) atomic_add_f32(&deg[(int)rows[e]], 1.0f);
}

// 2) dinv = rsqrt(deg) in place (deg >= 1 always due to self-loops)
__global__ void dinv_kernel(float* __restrict__ deg) {
  int i = blockIdx.x * blockDim.x + threadIdx.x;
  if (i < N_NODES) deg[i] = rsqrtf(deg[i]);
}

// 3) H = X @ W1  via chained V_WMMA_F32_16X16X4_F32; one 16x16 tile per wave.
__global__ void gemm1_wmma(const float* __restrict__ X,
                           const float* __restrict__ W,
                           float* __restrict__ H) {
  const int lane = threadIdx.x & 31;
  const int wave = threadIdx.x >> 5;
  const int gw   = blockIdx.x * 8 + wave;          // 8 waves / 256-thread block
  const int tilesN = HIDDEN / 16;                  // 4
  const int tm = gw / tilesN, tn = gw % tilesN;
  if (tm >= N_NODES / 16) return;                  // wave-uniform guard
  const int m0 = tm * 16, n0 = tn * 16;
  const int half = lane >> 4;                      // 0: lanes 0-15, 1: lanes 16-31
  const int l    = lane & 15;

  v8f c = {};
#pragma unroll
  for (int k0 = 0; k0 < F_IN; k0 += 4) {
    const int ka = k0 + half * 2;                  // ISA A/B 16x4 f32 layout
    v2f a, b;
    a.x = X[(m0 + l) * F_IN + ka];
    a.y = X[(m0 + l) * F_IN + ka + 1];
    b.x = W[ka * HIDDEN + (n0 + l)];
    b.y = W[(ka + 1) * HIDDEN + (n0 + l)];
    c = wmma_f32x4(a, b, c);
  }
#pragma unroll
  for (int j = 0; j < 8; ++j) {                    // D: VGPR j -> M = j + 8*half
    const int m = m0 + j + half * 8;
    H[m * HIDDEN + n0 + l] = c[j];
  }
}

// 4) scatter layer 1: items [0,E) are edges, [E,E+N) self-loops. 16 lanes/item.
__global__ void scatter1(const long long* __restrict__ rows,
                         const long long* __restrict__ cols,
                         const float* __restrict__ dinv,
                         const float* __restrict__ H,
                         float* __restrict__ agg) {
  const int t    = blockIdx.x * blockDim.x + threadIdx.x;
  const int item = t >> 4;
  const int sub  = t & 15;
  if (item >= N_EDGES + N_NODES) return;
  int r, c;
  if (item < N_EDGES) { r = (int)rows[item]; c = (int)cols[item]; }
  else                { r = c = item - N_EDGES; }
  const float nrm = dinv[r] * dinv[c];
  const float4 v = *(const float4*)(H + c * HIDDEN + sub * 4);
  float* dst = agg + r * HIDDEN + sub * 4;
  atomic_add_f32(dst + 0, v.x * nrm);
  atomic_add_f32(dst + 1, v.y * nrm);
  atomic_add_f32(dst + 2, v.z * nrm);
  atomic_add_f32(dst + 3, v.w * nrm);
}

// 5) h1 = relu(agg1 + b1) in place (bias applied AFTER aggregation, per reference)
__global__ void relu_bias(float* __restrict__ agg, const float* __restrict__ b1) {
  int i = blockIdx.x * blockDim.x + threadIdx.x;
  if (i < N_NODES * HIDDEN) {
    float v = agg[i] + b1[i & (HIDDEN - 1)];
    agg[i] = v > 0.0f ? v : 0.0f;
  }
}

// 6) Z = H1 @ W2 (64x2), WMMA with B zero-padded to 16 cols; store cols 0..1.
__global__ void gemm2_wmma(const float* __restrict__ H1,
                           const float* __restrict__ W2,
                           float* __restrict__ Z) {
  const int lane = threadIdx.x & 31;
  const int wave = threadIdx.x >> 5;
  const int gw   = blockIdx.x * 8 + wave;
  if (gw >= N_NODES / 16) return;                  // wave-uniform guard
  const int m0 = gw * 16;
  const int half = lane >> 4;
  const int l    = lane & 15;

  v8f c = {};
#pragma unroll
  for (int k0 = 0; k0 < HIDDEN; k0 += 4) {
    const int ka = k0 + half * 2;
    v2f a, b;
    a.x = H1[(m0 + l) * HIDDEN + ka];
    a.y = H1[(m0 + l) * HIDDEN + ka + 1];
    b.x = (l < N_CLASSES) ? W2[ka * N_CLASSES + l] : 0.0f;
    b.y = (l < N_CLASSES) ? W2[(ka + 1) * N_CLASSES + l] : 0.0f;
    c = wmma_f32x4(a, b, c);
  }
  if (l < N_CLASSES) {                             // divergence only after WMMAs
#pragma unroll
    for (int j = 0; j < 8; ++j) {
      const int m = m0 + j + half * 8;
      Z[m * N_CLASSES + l] = c[j];
    }
  }
}

// 7) scatter layer 2: one thread per item, 2 features.
__global__ void scatter2(const long long* __restrict__ rows,
                         const long long* __restrict__ cols,
                         const float* __restrict__ dinv,
                         const float* __restrict__ Z,
                         float* __restrict__ agg2) {
  const int item = blockIdx.x * blockDim.x + threadIdx.x;
  if (item >= N_EDGES + N_NODES) return;
  int r, c;
  if (item < N_EDGES) { r = (int)rows[item]; c = (int)cols[item]; }
  else                { r = c = item - N_EDGES; }
  const float nrm = dinv[r] * dinv[c];
  const float2 z = *(const float2*)(Z + c * N_CLASSES);
  atomic_add_f32(agg2 + r * N_CLASSES + 0, z.x * nrm);
  atomic_add_f32(agg2 + r * N_CLASSES + 1, z.y * nrm);
}

// 8) logits = agg2 + b2; 2-class softmax -> d_out
__global__ void final_softmax(const float* __restrict__ agg2,
                              const float* __restrict__ b2,
                              float* __restrict__ out) {
  int i = blockIdx.x * blockDim.x + threadIdx.x;
  if (i < N_NODES) {
    float l0 = agg2[i * 2 + 0] + b2[0];
    float l1 = agg2[i * 2 + 1] + b2[1];
    float mx = fmaxf(l0, l1);
    float e0 = expf(l0 - mx), e1 = expf(l1 - mx);
    float inv = 1.0f / (e0 + e1);
    out[i * 2 + 0] = e0 * inv;
    out[i * 2 + 1] = e1 * inv;
  }
}

extern "C" void kernel_launch(void* const* d_in, const int* in_sizes, int n_in,
                              void* d_out, int out_size, void* d_ws, size_t ws_size,
                              hipStream_t stream) {
  (void)in_sizes; (void)n_in; (void)out_size; (void)ws_size;
  const float*     x     = (const float*)d_in[0];
  const long long* ei    = (const long long*)d_in[1];   // int64 [2, E]
  const float*     W1    = (const float*)d_in[2];
  const float*     b1    = (const float*)d_in[3];
  const float*     W2    = (const float*)d_in[4];
  const float*     b2    = (const float*)d_in[5];
  const long long* rows  = ei;                          // edge_index[0] (targets)
  const long long* cols  = ei + N_EDGES;                // edge_index[1] (sources)
  float* out = (float*)d_out;

  // workspace layout (floats): deg/dinv | H | agg1(->h1) | Z | agg2  (~53 MB)
  float* ws   = (float*)d_ws;
  float* deg  = ws;
  float* H    = deg + N_NODES;
  float* agg1 = H + (size_t)N_NODES * HIDDEN;
  float* Z    = agg1 + (size_t)N_NODES * HIDDEN;
  float* agg2 = Z + (size_t)N_NODES * N_CLASSES;

  const int B = 256;
  init_ws<<<(N_NODES * HIDDEN + B - 1) / B, B, 0, stream>>>(deg, agg1, agg2);
  degree_kernel<<<(N_EDGES + B - 1) / B, B, 0, stream>>>(rows, deg);
  dinv_kernel<<<(N_NODES + B - 1) / B, B, 0, stream>>>(deg);

  // Layer 1
  gemm1_wmma<<<((N_NODES / 16) * (HIDDEN / 16)) / 8, B, 0, stream>>>(x, W1, H);
  scatter1<<<((N_EDGES + N_NODES) * 16 + B - 1) / B, B, 0, stream>>>(rows, cols, deg, H, agg1);
  relu_bias<<<(N_NODES * HIDDEN + B - 1) / B, B, 0, stream>>>(agg1, b1);

  // Layer 2
  gemm2_wmma<<<((N_NODES / 16) + 7) / 8, B, 0, stream>>>(agg1, W2, Z);
  scatter2<<<(N_EDGES + N_NODES + B - 1) / B, B, 0, stream>>>(rows, cols, deg, Z, agg2);
  final_softmax<<<(N_NODES + B - 1) / B, B, 0, stream>>>(agg2, b2, out);
}